// ConversationalGraph_3676492005628
// MI455X (gfx1250) — compile-verified
//
#include <hip/hip_runtime.h>

typedef __attribute__((ext_vector_type(2))) float v2f;
typedef __attribute__((ext_vector_type(8))) float v8f;

#define LEAKY 0.01f
#define NUM_GRAPHS 512

// ---------------------------------------------------------------- utilities
__global__ void zero_f32(float* __restrict__ p, long long n) {
  long long i = (long long)blockIdx.x * blockDim.x + threadIdx.x;
  if (i < n) p[i] = 0.0f;
}

// ------------------------------------------------- edge gather/scale/scatter
// 16 threads per edge, each handles 4 contiguous features (float4 gather from
// L2-resident x, 4x global_atomic_add_f32 scatter).
__global__ void scatter_edges(const float* __restrict__ x,
                              const long long* __restrict__ src,
                              const long long* __restrict__ dst,
                              const float* __restrict__ w,
                              float* __restrict__ agg, long long E) {
  long long tid = (long long)blockIdx.x * blockDim.x + threadIdx.x;
  long long e = tid >> 4;
  if (e >= E) return;
  int q = (int)(tid & 15) * 4;
  long long s = src[e], d = dst[e];
  float wv = w[e];
  const float4 v = *(const float4*)(x + s * 64 + q);
  float* dp = agg + d * 64 + q;
  atomicAdd(dp + 0, v.x * wv);
  atomicAdd(dp + 1, v.y * wv);
  atomicAdd(dp + 2, v.z * wv);
  atomicAdd(dp + 3, v.w * wv);
}

__global__ void count_nodes(const long long* __restrict__ batch,
                            float* __restrict__ counts, int N) {
  int i = blockIdx.x * blockDim.x + threadIdx.x;
  if (i < N) atomicAdd(&counts[batch[i]], 1.0f);
}

// ---------------------------------------------------------- fused GraphConv
// out[n,:] = leaky( agg[n,:] @ Wrel + root[n,:] @ Wroot + b )   (K=128 fused)
// One wave32 = 16 node rows x 64 cols via V_WMMA_F32_16X16X4_F32.
// mode 0: store to out.  mode 1: atomically accumulate into per-graph pool.
__global__ __launch_bounds__(256) void gconv_wmma(
    const float* __restrict__ Arel, const float* __restrict__ Aroot,
    const float* __restrict__ Wrel, const float* __restrict__ Wroot,
    const float* __restrict__ bias,
    float* __restrict__ out, float* __restrict__ pooled,
    const long long* __restrict__ batch, int N, int mode) {
  // LDS: concatenated weights [128 x 64] stored as adjacent k-pairs so a
  // B fragment (B[k][n], B[k+1][n]) is one ds_load_b64.
  __shared__ v2f sW[64 * 64];  // 32 KB
  float* sWf = (float*)sW;
  int t = threadIdx.x;
  for (int i = t; i < 4096; i += 256) {
    int k = i >> 6, n = i & 63;
    sWf[((k >> 1) * 64 + n) * 2 + (k & 1)] = Wrel[i];          // rows 0..63
    sWf[(((k >> 1) + 32) * 64 + n) * 2 + (k & 1)] = Wroot[i];  // rows 64..127
  }
  __syncthreads();

  int wave = t >> 5, lane = t & 31;
  long long row0 = ((long long)blockIdx.x * 8 + wave) * 16;
  if (row0 >= N) return;  // whole wave exits; active waves have EXEC all-1s

  int mrow = lane & 15;   // A row / B-C column-within-tile
  int khalf = lane >> 4;  // 0: K pair {0,1}, 1: K pair {2,3} per step
  const float* ar = Arel + (row0 + mrow) * 64 + khalf * 2;
  const float* ax = Aroot + (row0 + mrow) * 64 + khalf * 2;

  v8f acc0 = {}, acc1 = {}, acc2 = {}, acc3 = {};

#define WSTEP(APTR, KPBASE, S)                                                 \
  do {                                                                         \
    v2f a = *(const v2f*)((APTR) + 4 * (S));                                   \
    int kp = (KPBASE) + 2 * (S) + khalf;                                       \
    v2f b0 = sW[kp * 64 + mrow + 0];                                           \
    v2f b1 = sW[kp * 64 + mrow + 16];                                          \
    v2f b2 = sW[kp * 64 + mrow + 32];                                          \
    v2f b3 = sW[kp * 64 + mrow + 48];                                          \
    acc0 = __builtin_amdgcn_wmma_f32_16x16x4_f32(false, a, false, b0,          \
                                                 (short)0, acc0, false, false);\
    acc1 = __builtin_amdgcn_wmma_f32_16x16x4_f32(false, a, false, b1,          \
                                                 (short)0, acc1, false, false);\
    acc2 = __builtin_amdgcn_wmma_f32_16x16x4_f32(false, a, false, b2,          \
                                                 (short)0, acc2, false, false);\
    acc3 = __builtin_amdgcn_wmma_f32_16x16x4_f32(false, a, false, b3,          \
                                                 (short)0, acc3, false, false);\
  } while (0)

#pragma unroll
  for (int s = 0; s < 16; ++s) WSTEP(ar, 0, s);   // agg @ Wrel   (K 0..63)
#pragma unroll
  for (int s = 0; s < 16; ++s) WSTEP(ax, 32, s);  // root @ Wroot (K 64..127)
#undef WSTEP

  // Epilogue. C/D layout: VGPR r -> M = r + 8*khalf, N = lane&15 within tile.
  int mhi = khalf * 8;
#define EPILOG(ACC, J)                                                         \
  do {                                                                         \
    int col = (J)*16 + mrow;                                                   \
    float bv = bias[col];                                                      \
    _Pragma("unroll") for (int r = 0; r < 8; ++r) {                            \
      long long row = row0 + mhi + r;                                          \
      float v = (ACC)[r] + bv;                                                 \
      v = (v > 0.0f) ? v : LEAKY * v;                                          \
      if (mode == 0) out[row * 64 + col] = v;                                  \
      else atomicAdd(&pooled[batch[row] * 64 + col], v);                       \
    }                                                                          \
  } while (0)
  EPILOG(acc0, 0);
  EPILOG(acc1, 1);
  EPILOG(acc2, 2);
  EPILOG(acc3, 3);
#undef EPILOG
}

// ------------------------------------------------------------- linear head
// out[g,c] = (pool[g,:]/max(count,1)) @ Wl + bl   (512x8, trivial)
__global__ void head_kernel(const float* __restrict__ pooled,
                            const float* __restrict__ counts,
                            const float* __restrict__ Wl,
                            const float* __restrict__ bl,
                            float* __restrict__ out) {
  int i = blockIdx.x * blockDim.x + threadIdx.x;  // 4096
  if (i >= NUM_GRAPHS * 8) return;
  int g = i >> 3, c = i & 7;
  float inv = 1.0f / fmaxf(counts[g], 1.0f);
  float s = 0.0f;
#pragma unroll 8
  for (int h = 0; h < 64; ++h) s += pooled[g * 64 + h] * Wl[h * 8 + c];
  out[i] = s * inv + bl[c];
}

// ------------------------------------------------------------------ launch
extern "C" void kernel_launch(void* const* d_in, const int* in_sizes, int n_in,
                              void* d_out, int out_size, void* d_ws,
                              size_t ws_size, hipStream_t stream) {
  const float* x = (const float*)d_in[0];
  const long long* ei = (const long long*)d_in[1];
  const float* w = (const float*)d_in[2];
  const long long* batch = (const long long*)d_in[3];
  const float* W1_root = (const float*)d_in[4];
  const float* W1_rel = (const float*)d_in[5];
  const float* b1 = (const float*)d_in[6];
  const float* W2_root = (const float*)d_in[7];
  const float* W2_rel = (const float*)d_in[8];
  const float* b2 = (const float*)d_in[9];
  const float* Wl = (const float*)d_in[10];
  const float* bl = (const float*)d_in[11];

  long long E = in_sizes[2];  // weights: [E]
  long long N = in_sizes[3];  // batch:   [N]
  const long long* src = ei;
  const long long* dst = ei + E;

  // workspace: agg[N*64] | h1[N*64] | pooled[512*64] | counts[512]
  float* agg = (float*)d_ws;
  float* h1 = agg + N * 64;
  float* pooled = h1 + N * 64;
  float* counts = pooled + NUM_GRAPHS * 64;

  long long aggN = N * 64;
  long long poolN = NUM_GRAPHS * 64 + NUM_GRAPHS;

  dim3 blk(256);
  int gZeroAgg = (int)((aggN + 255) / 256);
  int gZeroPool = (int)((poolN + 255) / 256);
  int gScatter = (int)((E * 16 + 255) / 256);
  int gCount = (int)((N + 255) / 256);
  int gGemm = (int)((N + 127) / 128);  // 8 waves/block, 16 rows/wave
  int gHead = (NUM_GRAPHS * 8 + 255) / 256;

  zero_f32<<<gZeroAgg, blk, 0, stream>>>(agg, aggN);
  zero_f32<<<gZeroPool, blk, 0, stream>>>(pooled, poolN);

  // ---- layer 1
  scatter_edges<<<gScatter, blk, 0, stream>>>(x, src, dst, w, agg, E);
  count_nodes<<<gCount, blk, 0, stream>>>(batch, counts, (int)N);
  gconv_wmma<<<gGemm, blk, 0, stream>>>(agg, x, W1_rel, W1_root, b1, h1,
                                        pooled, batch, (int)N, 0);

  // ---- layer 2 (pooling fused into epilogue; h2 never materialized)
  zero_f32<<<gZeroAgg, blk, 0, stream>>>(agg, aggN);
  scatter_edges<<<gScatter, blk, 0, stream>>>(h1, src, dst, w, agg, E);
  gconv_wmma<<<gGemm, blk, 0, stream>>>(agg, h1, W2_rel, W2_root, b2, h1,
                                        pooled, batch, (int)N, 1);

  // ---- mean + linear head
  head_kernel<<<gHead, blk, 0, stream>>>(pooled, counts, Wl, bl,
                                         (float*)d_out);
}